// DetectionLoss_41618233098445
// MI455X (gfx1250) — compile-verified
//
#include <hip/hip_runtime.h>
#include <math.h>

typedef __attribute__((ext_vector_type(2))) float v2f;
typedef __attribute__((ext_vector_type(8))) float v8f;

#define EPS_IOU 1e-6f
#define EPS_BCE 1e-7f
#define NBLOCKS 2048
#define TPB 256
#define NWAVES (TPB / 32)

// Full 32-lane wave sum via V_WMMA_F32_16X16X4_F32 with an all-ones B matrix.
// A layout (ISA 7.12.2, 32-bit A 16x4): lanes 0-15 -> M=lane, VGPR0=K0, VGPR1=K1;
// lanes 16-31 -> M=lane-16, VGPR0=K2, VGPR1=K3.  With a = {v, 0} and B == 1:
//   D[m,n] = v(m) + v(m+16)  (= s_m, constant across n)
// D layout: VGPR j holds M=j for lanes 0-15 and M=8+j for lanes 16-31, so
//   sum_j d[j] = s_0..s_7 (lanes 0-15) or s_8..s_15 (lanes 16-31);
// one xor-16 shuffle combines them into the full 32-lane sum in every lane.
__device__ __forceinline__ float wave_sum_wmma(float v) {
  v2f a; a.x = v;    a.y = 0.0f;
  v2f b; b.x = 1.0f; b.y = 1.0f;
  v8f c = {};
  c = __builtin_amdgcn_wmma_f32_16x16x4_f32(
      /*neg_a=*/false, a, /*neg_b=*/false, b,
      /*c_mod=*/(short)0, c, /*reuse_a=*/false, /*reuse_b=*/false);
  float t = c.s0 + c.s1 + c.s2 + c.s3 + c.s4 + c.s5 + c.s6 + c.s7;
  t += __shfl_xor(t, 16, 32);
  return t;
}

// Pass 1: streaming pass over all rows; per-block partial sums into ws.
__global__ void __launch_bounds__(TPB)
det_loss_partial(const float4* __restrict__ pred_coords,
                 const float*  __restrict__ pred_conf,
                 const float4* __restrict__ true_coords,
                 float* __restrict__ ws, int nrows) {
  __shared__ float sC[NWAVES];
  __shared__ float sB[NWAVES];

  const int tid    = blockIdx.x * blockDim.x + threadIdx.x;
  const int stride = gridDim.x * blockDim.x;

  float cs = 0.0f;  // sum of squared coord diffs (4 per row)
  float bs = 0.0f;  // sum of per-row BCE terms

  for (int i = tid; i < nrows; i += stride) {
    const float4 p = pred_coords[i];   // global_load_b128
    const float4 t = true_coords[i];   // global_load_b128
    const float pc = pred_conf[i];

    // MSE contribution
    const float dx = p.x - t.x, dy = p.y - t.y;
    const float dw = p.z - t.z, dh = p.w - t.w;
    cs += dx * dx + dy * dy + dw * dw + dh * dh;

    // IoU (same formula structure as the reference)
    const float px1 = p.x - 0.5f * p.z, px2 = p.x + 0.5f * p.z;
    const float py1 = p.y - 0.5f * p.w, py2 = p.y + 0.5f * p.w;
    const float tx1 = t.x - 0.5f * t.z, tx2 = t.x + 0.5f * t.z;
    const float ty1 = t.y - 0.5f * t.w, ty2 = t.y + 0.5f * t.w;
    const float x1 = fmaxf(px1, tx1), y1 = fmaxf(py1, ty1);
    const float x2 = fminf(px2, tx2), y2 = fminf(py2, ty2);
    const float inter = fmaxf(x2 - x1, 0.0f) * fmaxf(y2 - y1, 0.0f);
    const float pa = (px2 - px1) * (py2 - py1);
    const float ta = (tx2 - tx1) * (ty2 - ty1);
    const float iou = inter / (pa + ta - inter + EPS_IOU);

    // BCE(pred_conf, iou) with clamp
    const float pcl = fminf(fmaxf(pc, EPS_BCE), 1.0f - EPS_BCE);
    bs -= iou * logf(pcl) + (1.0f - iou) * log1pf(-pcl);
  }

  // wave reduction (WMMA), then cross-wave via LDS
  const float wc = wave_sum_wmma(cs);
  const float wb = wave_sum_wmma(bs);
  const int wave = threadIdx.x >> 5;
  const int lane = threadIdx.x & 31;
  if (lane == 0) { sC[wave] = wc; sB[wave] = wb; }
  __syncthreads();
  if (threadIdx.x == 0) {
    float c = 0.0f, b = 0.0f;
    #pragma unroll
    for (int w = 0; w < NWAVES; ++w) { c += sC[w]; b += sB[w]; }
    ws[2 * blockIdx.x + 0] = c;
    ws[2 * blockIdx.x + 1] = b;
  }
}

// Pass 2: one block reduces the per-block partials and writes the 3 scalars.
__global__ void __launch_bounds__(TPB)
det_loss_final(const float* __restrict__ ws, float* __restrict__ out,
               int nparts, int nrows) {
  __shared__ float sC[NWAVES];
  __shared__ float sB[NWAVES];

  float cs = 0.0f, bs = 0.0f;
  for (int i = threadIdx.x; i < nparts; i += blockDim.x) {
    cs += ws[2 * i + 0];
    bs += ws[2 * i + 1];
  }

  const float wc = wave_sum_wmma(cs);
  const float wb = wave_sum_wmma(bs);
  const int wave = threadIdx.x >> 5;
  const int lane = threadIdx.x & 31;
  if (lane == 0) { sC[wave] = wc; sB[wave] = wb; }
  __syncthreads();
  if (threadIdx.x == 0) {
    float c = 0.0f, b = 0.0f;
    #pragma unroll
    for (int w = 0; w < NWAVES; ++w) { c += sC[w]; b += sB[w]; }
    const float coord_loss = c / (4.0f * (float)nrows);
    const float conf_loss  = b / (float)nrows;
    out[0] = coord_loss + conf_loss;  // total (COORD_W = CONF_W = 1)
    out[1] = coord_loss;
    out[2] = conf_loss;
  }
}

extern "C" void kernel_launch(void* const* d_in, const int* in_sizes, int n_in,
                              void* d_out, int out_size, void* d_ws, size_t ws_size,
                              hipStream_t stream) {
  const float4* pred_coords = (const float4*)d_in[0];  // [B,4] f32
  const float*  pred_conf   = (const float*)d_in[1];   // [B,1] f32
  const float4* true_coords = (const float4*)d_in[2];  // [B,4] f32
  float* out = (float*)d_out;
  float* ws  = (float*)d_ws;

  const int nrows = in_sizes[0] / 4;

  det_loss_partial<<<NBLOCKS, TPB, 0, stream>>>(pred_coords, pred_conf,
                                                true_coords, ws, nrows);
  det_loss_final<<<1, TPB, 0, stream>>>(ws, out, NBLOCKS, nrows);
}